// ReasoningDiversityLoss_32263794327911
// MI455X (gfx1250) — compile-verified
//
#include <hip/hip_runtime.h>
#include <math.h>

// ---------------- Problem constants (from reference) ----------------
#define B_CHAINS 64
#define S_STEPS  1024
#define D_DIM    256
#define N_NEG    32
#define MARGIN   1.0f
#define DIV_W    0.1f
#define EPS_REF  1e-8f

#define BLOCKS_PER_CHAIN 16
#define ROWS_PER_BLOCK   (S_STEPS / BLOCKS_PER_CHAIN)   // 64
#define WAVES_PER_BLOCK  8
#define ROWS_PER_WAVE    (ROWS_PER_BLOCK / WAVES_PER_BLOCK) // 8

typedef float v2f __attribute__((ext_vector_type(2)));
typedef float v8f __attribute__((ext_vector_type(8)));

__device__ __forceinline__ float waveReduceSum(float v) {
#pragma unroll
  for (int m = 16; m >= 1; m >>= 1) v += __shfl_xor(v, m, 32);
  return v;
}

// ---------------- Kernel 0: zero workspace accumulators ----------------
__global__ void rdl_zero(float* __restrict__ ws, int n) {
  int i = blockIdx.x * blockDim.x + threadIdx.x;
  if (i < n) ws[i] = 0.0f;
}

// ---------------- Kernel 1: single pass over x ----------------
// Accumulates chain_sum[b][d] = sum_s normalize(x[b,s,:])[d]  and
// psum[b] = sum over (s,d) of x[b,s,d].
__global__ void rdl_pass1(const float* __restrict__ x,
                          float* __restrict__ chain_sum,
                          float* __restrict__ psum) {
  const int blk   = blockIdx.x;                 // 0..1023
  const int chain = blk / BLOCKS_PER_CHAIN;
  const int chunk = blk % BLOCKS_PER_CHAIN;
  const int tid   = threadIdx.x;
  const int wave  = tid >> 5;
  const int lane  = tid & 31;

  __shared__ float sAcc[D_DIM];
  __shared__ float sRaw;
  for (int i = tid; i < D_DIM; i += blockDim.x) sAcc[i] = 0.0f;
  if (tid == 0) sRaw = 0.0f;
  __syncthreads();

  const float* base =
      x + ((size_t)chain * S_STEPS + (size_t)chunk * ROWS_PER_BLOCK) * D_DIM;

  float acc0 = 0.f, acc1 = 0.f, acc2 = 0.f, acc3 = 0.f;
  float acc4 = 0.f, acc5 = 0.f, acc6 = 0.f, acc7 = 0.f;
  float raw  = 0.f;

#pragma unroll
  for (int r = 0; r < ROWS_PER_WAVE; ++r) {
    const int row = wave * ROWS_PER_WAVE + r;
    const float4* rp = (const float4*)(base + (size_t)row * D_DIM);
    float4 a = rp[lane];        // elements 4*lane .. 4*lane+3
    float4 b = rp[32 + lane];   // elements 128+4*lane .. 128+4*lane+3
    float ss = a.x*a.x + a.y*a.y + a.z*a.z + a.w*a.w
             + b.x*b.x + b.y*b.y + b.z*b.z + b.w*b.w;
    ss = waveReduceSum(ss);
    const float inv = 1.0f / fmaxf(sqrtf(ss), EPS_REF);
    acc0 += a.x * inv; acc1 += a.y * inv; acc2 += a.z * inv; acc3 += a.w * inv;
    acc4 += b.x * inv; acc5 += b.y * inv; acc6 += b.z * inv; acc7 += b.w * inv;
    raw  += a.x + a.y + a.z + a.w + b.x + b.y + b.z + b.w;
  }

  const int e0 = lane * 4;
  const int e1 = 128 + lane * 4;
  atomicAdd(&sAcc[e0 + 0], acc0);
  atomicAdd(&sAcc[e0 + 1], acc1);
  atomicAdd(&sAcc[e0 + 2], acc2);
  atomicAdd(&sAcc[e0 + 3], acc3);
  atomicAdd(&sAcc[e1 + 0], acc4);
  atomicAdd(&sAcc[e1 + 1], acc5);
  atomicAdd(&sAcc[e1 + 2], acc6);
  atomicAdd(&sAcc[e1 + 3], acc7);

  raw = waveReduceSum(raw);
  if (lane == 0) atomicAdd(&sRaw, raw);
  __syncthreads();

  for (int i = tid; i < D_DIM; i += blockDim.x)
    atomicAdd(&chain_sum[chain * D_DIM + i], sAcc[i]);
  if (tid == 0) atomicAdd(&psum[chain], sRaw);
}

// ---------------- Kernel 2: finalize (f32 WMMA Gram + scalars) ----------------
// One block, 512 threads = 16 waves. Wave w computes Gram tile (w/4, w%4)
// of G = chain_sum . chain_sum^T using V_WMMA_F32_16X16X4_F32, K=256.
__global__ void rdl_finalize(const float* __restrict__ chain_sum,
                             const float* __restrict__ psum,
                             const float* __restrict__ negs,
                             float* __restrict__ out) {
  const int tid  = threadIdx.x;
  const int wave = tid >> 5;   // 0..15
  const int lane = tid & 31;

  __shared__ float sSumSq[B_CHAINS];
  __shared__ float sN[N_NEG];
  __shared__ float sInter;
  __shared__ float sHard;
  __shared__ float sIntra;

  if (tid == 0) { sInter = 0.f; sHard = 0.f; sIntra = 0.f; }
  __syncthreads();

  // ||chain_sum[b]||^2, 4 chains per wave
  for (int b = wave; b < B_CHAINS; b += 16) {
    const float* row = chain_sum + b * D_DIM;
    float s = 0.f;
    for (int i = lane; i < D_DIM; i += 32) { float v = row[i]; s += v * v; }
    s = waveReduceSum(s);
    if (lane == 0) sSumSq[b] = s;
  }

  // mean of each hard-negative row
  for (int k = wave; k < N_NEG; k += 16) {
    const float* row = negs + k * D_DIM;
    float s = 0.f;
    for (int i = lane; i < D_DIM; i += 32) s += row[i];
    s = waveReduceSum(s);
    if (lane == 0) sN[k] = s * (1.0f / (float)D_DIM);
  }

  // ---- Gram tile via f32 WMMA ----
  // A (16x4 f32): lanes 0-15 hold M rows; v0 = K0 (lo half) / K2 (hi half),
  // v1 = K1 / K3. B (4x16): N on lanes, K on {VGPR, half} -> for symmetric
  // G = M.M^T both fragments load identically from chain_sum.
  const int ti = wave >> 2, tj = wave & 3;
  const int hi = lane >> 4;
  const int mr = lane & 15;
  const float* pa = chain_sum + (ti * 16 + mr) * D_DIM + 2 * hi;
  const float* pb = chain_sum + (tj * 16 + mr) * D_DIM + 2 * hi;

  v8f c = {};
#pragma unroll 4
  for (int k0 = 0; k0 < D_DIM; k0 += 4) {
    v2f a, b;
    a.x = pa[k0]; a.y = pa[k0 + 1];
    b.x = pb[k0]; b.y = pb[k0 + 1];
    c = __builtin_amdgcn_wmma_f32_16x16x4_f32(
        /*neg_a=*/false, a, /*neg_b=*/false, b,
        /*c_mod=*/(short)0, c, /*reuse_a=*/false, /*reuse_b=*/false);
  }

  // C/D layout: VGPR r -> M = r (lanes 0-15) or 8+r (lanes 16-31); N = lane%16
  const float invS2 = 1.0f / ((float)S_STEPS * (float)S_STEPS);
  float interPart = 0.f;
#pragma unroll
  for (int r = 0; r < 8; ++r) {
    const int grow = ti * 16 + r + 8 * hi;
    const int gcol = tj * 16 + mr;
    if (grow < gcol) interPart += fminf(c[r] * invS2, MARGIN);
  }
  interPart = waveReduceSum(interPart);
  if (lane == 0) atomicAdd(&sInter, interPart);
  __syncthreads();

  // intra: mean_b min((||chain_sum||^2 - S) / (S(S-1)), MARGIN)
  if (tid < B_CHAINS) {
    const float ia =
        (sSumSq[tid] - (float)S_STEPS) /
        ((float)S_STEPS * (float)(S_STEPS - 1));
    atomicAdd(&sIntra, fminf(ia, MARGIN) * (1.0f / (float)B_CHAINS));
  }

  // hard: mean over (b,k) of relu(n[k] - p[b] + MARGIN)
  const float invSD = 1.0f / ((float)S_STEPS * (float)D_DIM);
  float h = 0.f;
  for (int i = tid; i < B_CHAINS * N_NEG; i += 512) {
    const int b = i >> 5;       // /32
    const int k = i & 31;
    const float p = psum[b] * invSD;
    h += fmaxf(sN[k] - p + MARGIN, 0.0f);
  }
  h = waveReduceSum(h);
  if (lane == 0) atomicAdd(&sHard, h);
  __syncthreads();

  if (tid == 0) {
    const float intra = sIntra;
    const float inter = sInter * (2.0f / ((float)B_CHAINS * (B_CHAINS - 1)));
    const float hard  = sHard * (1.0f / ((float)B_CHAINS * (float)N_NEG));
    out[0] = intra;
    out[1] = inter;
    out[2] = hard;
    out[3] = intra + inter + DIV_W * hard;
  }
}

// ---------------- Host launcher ----------------
extern "C" void kernel_launch(void* const* d_in, const int* in_sizes, int n_in,
                              void* d_out, int out_size, void* d_ws, size_t ws_size,
                              hipStream_t stream) {
  (void)in_sizes; (void)n_in; (void)out_size; (void)ws_size;
  const float* x    = (const float*)d_in[0];   // [64,1024,256] f32
  const float* negs = (const float*)d_in[1];   // [32,256] f32
  float* out = (float*)d_out;                  // 4 floats

  float* chain_sum = (float*)d_ws;                  // 64*256
  float* psum      = chain_sum + B_CHAINS * D_DIM;  // 64

  const int n_zero = B_CHAINS * D_DIM + B_CHAINS;
  rdl_zero<<<(n_zero + 255) / 256, 256, 0, stream>>>((float*)d_ws, n_zero);

  rdl_pass1<<<B_CHAINS * BLOCKS_PER_CHAIN, WAVES_PER_BLOCK * 32, 0, stream>>>(
      x, chain_sum, psum);

  rdl_finalize<<<1, 512, 0, stream>>>(chain_sum, psum, negs, out);
}